// SAGE_74148315398477
// MI455X (gfx1250) — compile-verified
//
#include <hip/hip_runtime.h>
#include <hip/hip_bf16.h>
#include <math.h>

// ---------------------------------------------------------------------------
// 2-layer GraphSAGE forward for MI455X (gfx1250, wave32, WMMA, async-LDS).
//   layer0: h   = relu( mean_scatter(x[row0],col0,20000) @ Wl0 + bl0 + x[:20000] @ Wr0 )
//   layer1: hW  = h @ Wl1            (transform-first: mean(h)@Wl1 == mean(h@Wl1))
//           out = log_softmax( mean_scatter(hW[row1],col1,4000) + bl1 + h[:4000] @ Wr1 )
// GEMMs: V_WMMA_F32_16X16X4_F32 with W panels staged to LDS via
// GLOBAL_LOAD_ASYNC_TO_LDS_B128 (ASYNCcnt) and reused across 5 M-tiles/block.
// ---------------------------------------------------------------------------

#define N_SRC  100000
#define N_T0   20000
#define N_T1   4000
#define E0     1000000
#define E1     300000
#define D_IN   256
#define D_HID  256
#define D_OUT  64
#define MTILES 5          // M-tiles per block (panel reuse)

typedef __attribute__((ext_vector_type(2))) float v2f;
typedef __attribute__((ext_vector_type(8))) float v8f;
typedef __attribute__((address_space(3))) float lds_float;

// LDS byte offset of a pointer into (dynamic) shared memory
static __device__ __forceinline__ uint32_t lds_offset(const float* p) {
    return (uint32_t)(uintptr_t)(lds_float*)p;
}

// async 16B copy global -> LDS (tracked by ASYNCcnt)
static __device__ __forceinline__ void async_copy_f4(uint32_t lds_byte_off,
                                                     const float* gsrc) {
    asm volatile("global_load_async_to_lds_b128 %0, %1, off"
                 :: "v"(lds_byte_off), "v"(gsrc)
                 : "memory");
}

static __device__ __forceinline__ void async_wait_all() {
    asm volatile("s_wait_asynccnt 0x0" ::: "memory");
}

// ---------------------------------------------------------------- zero fill
__global__ void sage_zero_f32(float* __restrict__ p, size_t n) {
    size_t i      = (size_t)blockIdx.x * blockDim.x + threadIdx.x;
    size_t stride = (size_t)gridDim.x * blockDim.x;
    for (; i < n; i += stride) p[i] = 0.0f;
}

// ------------------------------------------------- edge scatter-add (D=256)
// One wave per edge: 2x float4 per lane (coalesced 1KB/row), 8 L2 atomics.
__global__ void sage_scatter_add256(const float* __restrict__ src,
                                    const int* __restrict__ row,
                                    const int* __restrict__ col,
                                    float* __restrict__ sums,
                                    float* __restrict__ cnt,
                                    int nedges) {
    int gtid = blockIdx.x * blockDim.x + threadIdx.x;
    int e    = gtid >> 5;
    int lane = gtid & 31;
    if (e >= nedges) return;
    int r = row[e];
    int c = col[e];
    const float4* s4 = (const float4*)(src + (size_t)r * 256);
    float4 v0 = s4[lane];
    float4 v1 = s4[lane + 32];
    float* dst = sums + (size_t)c * 256;
    int o0 = lane * 4;
    atomicAdd(dst + o0 + 0, v0.x);
    atomicAdd(dst + o0 + 1, v0.y);
    atomicAdd(dst + o0 + 2, v0.z);
    atomicAdd(dst + o0 + 3, v0.w);
    atomicAdd(dst + 128 + o0 + 0, v1.x);
    atomicAdd(dst + 128 + o0 + 1, v1.y);
    atomicAdd(dst + 128 + o0 + 2, v1.z);
    atomicAdd(dst + 128 + o0 + 3, v1.w);
    if (lane == 0) atomicAdd(cnt + c, 1.0f);
}

// -------------------------------------------------- edge scatter-add (D=64)
__global__ void sage_scatter_add64(const float* __restrict__ src,
                                   const int* __restrict__ row,
                                   const int* __restrict__ col,
                                   float* __restrict__ sums,
                                   float* __restrict__ cnt,
                                   int nedges) {
    int gtid = blockIdx.x * blockDim.x + threadIdx.x;
    int e    = gtid >> 5;
    int lane = gtid & 31;
    if (e >= nedges) return;
    int r = row[e];
    int c = col[e];
    float2 v = ((const float2*)(src + (size_t)r * 64))[lane];
    float* dst = sums + (size_t)c * 64;
    atomicAdd(dst + lane * 2 + 0, v.x);
    atomicAdd(dst + lane * 2 + 1, v.y);
    if (lane == 0) atomicAdd(cnt + c, 1.0f);
}

// ----------------------------------------- layer-0 fused pair GEMM (WMMA)
// out = relu( (sums/cnt) @ Wl + bias + xt @ Wr ), K=256, N arbitrary mult 64.
// Block: 128 thr (4 waves, 64 cols), panels Wl/Wr staged to LDS (2*K*64 f32),
// reused over MTILES M-tiles. Dyn-LDS = 2*K*64*4 bytes.
__global__ void sage_gemm_pair_wmma(const float* __restrict__ sums,
                                    const float* __restrict__ cnt,
                                    const float* __restrict__ xt,
                                    const float* __restrict__ Wl,
                                    const float* __restrict__ Wr,
                                    const float* __restrict__ bias,
                                    float* __restrict__ out,
                                    int K, int N) {
    extern __shared__ float lds[];
    float* WlS = lds;             // [K][64]
    float* WrS = lds + K * 64;
    const int tid   = threadIdx.x;
    const int n0blk = blockIdx.x * 64;

    // stage 64-column panels of Wl and Wr via async copy
    const int nvec = K * 16;      // float4s per panel
    for (int i = tid; i < nvec; i += blockDim.x) {
        int krow = i >> 4;
        int j4   = (i & 15) << 2;
        async_copy_f4(lds_offset(WlS + krow * 64 + j4),
                      Wl + (size_t)krow * N + n0blk + j4);
        async_copy_f4(lds_offset(WrS + krow * 64 + j4),
                      Wr + (size_t)krow * N + n0blk + j4);
    }
    async_wait_all();
    __syncthreads();

    const int lane  = tid & 31;
    const int wave  = tid >> 5;
    const int coll  = wave * 16 + (lane & 15);   // col within 64-wide panel
    const int colB  = n0blk + coll;
    const int khalf = (lane >> 4) * 2;           // f32 16x4 frag: hi half holds K+2,K+3
    const float b   = bias[colB];

    for (int t = 0; t < MTILES; ++t) {
        const int m0   = (blockIdx.y * MTILES + t) * 16;
        const int rowA = m0 + (lane & 15);
        const float invc = 1.0f / fmaxf(cnt[rowA], 1.0f);
        const float* arow = sums + (size_t)rowA * K;
        const float* xrow = xt   + (size_t)rowA * K;

        v8f acc = {};
        for (int k = 0; k < K; k += 4) {
            const int kk = k + khalf;
            float2 sa = *(const float2*)(arow + kk);
            float2 xa = *(const float2*)(xrow + kk);
            v2f a;  a.x  = sa.x * invc; a.y  = sa.y * invc;
            v2f ax; ax.x = xa.x;        ax.y = xa.y;
            v2f bL; bL.x = WlS[kk * 64 + coll]; bL.y = WlS[(kk + 1) * 64 + coll];
            v2f bR; bR.x = WrS[kk * 64 + coll]; bR.y = WrS[(kk + 1) * 64 + coll];
            acc = __builtin_amdgcn_wmma_f32_16x16x4_f32(false, a,  false, bL,
                                                        (short)0, acc, false, false);
            acc = __builtin_amdgcn_wmma_f32_16x16x4_f32(false, ax, false, bR,
                                                        (short)0, acc, false, false);
        }
        const int rbase = m0 + (lane >> 4) * 8;
        #pragma unroll
        for (int r = 0; r < 8; ++r) {
            float v = acc[r] + b;
            out[(size_t)(rbase + r) * N + colB] = fmaxf(v, 0.0f);
        }
    }
}

// --------------------------------------- single-chain GEMM (WMMA), N == 64
// out = xt @ W  [+ msums/cnt (per-row mean add)] [+ bias]
// Used for hW = h@Wl1 (msums=bias=null) and the final
// out = mean(hW) + bl1 + h[:4000]@Wr1 (msums=sums1).
__global__ void sage_gemm_single_wmma(const float* __restrict__ xt,
                                      const float* __restrict__ W,
                                      const float* __restrict__ msums,
                                      const float* __restrict__ cnt,
                                      const float* __restrict__ bias,
                                      float* __restrict__ out,
                                      int K) {
    extern __shared__ float lds[];
    float* WS = lds;              // [K][64]
    const int tid = threadIdx.x;
    const int N   = 64;

    const int nvec = K * 16;
    for (int i = tid; i < nvec; i += blockDim.x) {
        int krow = i >> 4;
        int j4   = (i & 15) << 2;
        async_copy_f4(lds_offset(WS + krow * 64 + j4),
                      W + (size_t)krow * N + j4);
    }
    async_wait_all();
    __syncthreads();

    const int lane  = tid & 31;
    const int wave  = tid >> 5;
    const int colB  = wave * 16 + (lane & 15);
    const int khalf = (lane >> 4) * 2;
    const float b   = bias ? bias[colB] : 0.0f;

    for (int t = 0; t < MTILES; ++t) {
        const int m0   = (blockIdx.y * MTILES + t) * 16;
        const int rowA = m0 + (lane & 15);
        const float* xrow = xt + (size_t)rowA * K;

        v8f acc = {};
        for (int k = 0; k < K; k += 4) {
            const int kk = k + khalf;
            float2 xa = *(const float2*)(xrow + kk);
            v2f ax; ax.x = xa.x; ax.y = xa.y;
            v2f bW; bW.x = WS[kk * 64 + colB]; bW.y = WS[(kk + 1) * 64 + colB];
            acc = __builtin_amdgcn_wmma_f32_16x16x4_f32(false, ax, false, bW,
                                                        (short)0, acc, false, false);
        }
        const int rbase = m0 + (lane >> 4) * 8;
        #pragma unroll
        for (int r = 0; r < 8; ++r) {
            const int rr = rbase + r;
            float v = acc[r] + b;
            if (msums) {
                float c = fmaxf(cnt[rr], 1.0f);
                v += msums[(size_t)rr * N + colB] * (1.0f / c);
            }
            out[(size_t)rr * N + colB] = v;
        }
    }
}

// ------------------------------------------------ log_softmax over 64 cols
__global__ void sage_log_softmax64(float* __restrict__ io, int nrows) {
    int wave = (blockIdx.x * blockDim.x + threadIdx.x) >> 5;
    int lane = threadIdx.x & 31;
    if (wave >= nrows) return;
    float* p = io + (size_t)wave * 64;
    float v0 = p[lane];
    float v1 = p[lane + 32];
    float m = fmaxf(v0, v1);
    #pragma unroll
    for (int off = 16; off > 0; off >>= 1) m = fmaxf(m, __shfl_xor(m, off, 32));
    float s = __expf(v0 - m) + __expf(v1 - m);
    #pragma unroll
    for (int off = 16; off > 0; off >>= 1) s += __shfl_xor(s, off, 32);
    float l = __logf(s);
    p[lane]      = v0 - m - l;
    p[lane + 32] = v1 - m - l;
}

// ---------------------------------------------------------------------------
extern "C" void kernel_launch(void* const* d_in, const int* in_sizes, int n_in,
                              void* d_out, int out_size, void* d_ws, size_t ws_size,
                              hipStream_t stream) {
    const float* x    = (const float*)d_in[0];
    const int*   row0 = (const int*)  d_in[1];
    const int*   col0 = (const int*)  d_in[2];
    const int*   row1 = (const int*)  d_in[3];
    const int*   col1 = (const int*)  d_in[4];
    const float* Wl0  = (const float*)d_in[5];
    const float* bl0  = (const float*)d_in[6];
    const float* Wr0  = (const float*)d_in[7];
    const float* Wl1  = (const float*)d_in[8];
    const float* bl1  = (const float*)d_in[9];
    const float* Wr1  = (const float*)d_in[10];
    float* out = (float*)d_out;

    // Workspace (floats): [sums0][cnt0][sums1(64w)][cnt1][h][hW]
    float* ws    = (float*)d_ws;
    float* sums0 = ws;
    float* cnt0  = sums0 + (size_t)N_T0 * D_HID;
    float* sums1 = cnt0  + N_T0;
    float* cnt1  = sums1 + (size_t)N_T1 * D_OUT;
    float* h     = cnt1  + N_T1;
    float* hW    = h     + (size_t)N_T0 * D_HID;
    const size_t zero_count =
        (size_t)N_T0 * D_HID + N_T0 + (size_t)N_T1 * D_OUT + N_T1;

    // 1) zero the scatter accumulators (contiguous prefix)
    sage_zero_f32<<<4096, 256, 0, stream>>>(ws, zero_count);

    // 2) layer-0 scatter-add (1M edges, wave/edge)
    sage_scatter_add256<<<(E0 * 32 + 255) / 256, 256, 0, stream>>>(
        x, row0, col0, sums0, cnt0, E0);

    // 3) layer-0 fused pair GEMM -> h   (M=20000: 1250 tiles = 250 blocks x 5)
    sage_gemm_pair_wmma<<<dim3(D_HID / 64, (N_T0 / 16) / MTILES), 128,
                          2 * D_IN * 64 * sizeof(float), stream>>>(
        sums0, cnt0, x, Wl0, Wr0, bl0, h, D_IN, D_HID);

    // 4) transform-first: hW = h @ Wl1   (M=20000, N=64)
    sage_gemm_single_wmma<<<dim3(1, (N_T0 / 16) / MTILES), 128,
                            D_HID * 64 * sizeof(float), stream>>>(
        h, Wl1, nullptr, nullptr, nullptr, hW, D_HID);

    // 5) layer-1 scatter-add over hW (64-wide: 4x less atomic/gather traffic)
    sage_scatter_add64<<<(E1 * 32 + 255) / 256, 256, 0, stream>>>(
        hW, row1, col1, sums1, cnt1, E1);

    // 6) final: out = mean(hW) + bl1 + h[:4000] @ Wr1   (M=4000, N=64)
    sage_gemm_single_wmma<<<dim3(1, (N_T1 / 16) / MTILES), 128,
                            D_HID * 64 * sizeof(float), stream>>>(
        h, Wr1, sums1, cnt1, bl1, out, D_HID);

    // 7) in-place log_softmax over the 64 classes
    sage_log_softmax64<<<(N_T1 * 32 + 127) / 128, 128, 0, stream>>>(out, N_T1);
}